// GraphEncoder_28939489640781
// MI455X (gfx1250) — compile-verified
//
#include <hip/hip_runtime.h>

#define D_FEAT 128

typedef __attribute__((ext_vector_type(2))) float v2f;
typedef __attribute__((ext_vector_type(8))) float v8f;

// ---------------------------------------------------------------- utilities
__global__ void k_fill_zero(float* __restrict__ p, int n) {
  int i = blockIdx.x * blockDim.x + threadIdx.x;
  if (i < n) p[i] = 0.0f;
}

__global__ void k_deg(const int* __restrict__ dst, float* __restrict__ deg, int E) {
  int e = blockIdx.x * blockDim.x + threadIdx.x;
  if (e < E) atomicAdd(&deg[dst[e]], 1.0f);
}

__global__ void k_dinv(float* __restrict__ deg, int N) {
  int i = blockIdx.x * blockDim.x + threadIdx.x;
  if (i < N) deg[i] = rsqrtf(deg[i] + 1.0f);   // deg counts dst edges; +1 self loop
}

// ---------------------------------------------------------------- WMMA GEMM
// H[M x 128] = X[M x 128] @ W[128 x 128], fp32 path (V_WMMA_F32_16X16X4_F32).
// One wave owns one 16x16 output tile; 8 waves/block tile the full N=128.
// A 16x4 layout : lanes 0-15 -> M=row, v0=K(2*kg+0), v1=K(2*kg+1), kg = lane>=16
// B 4x16 layout : lane&15 -> N col, v0=K(2*kg+0), v1=K(2*kg+1)
// C/D 16x16     : VGPR r -> M = r + 8*kg, lane&15 -> N col
__global__ void __launch_bounds__(256) k_gemm_wmma(const float* __restrict__ X,
                                                   const float* __restrict__ W,
                                                   float* __restrict__ H) {
  const int wave = threadIdx.x >> 5;
  const int lane = threadIdx.x & 31;
  const int row  = lane & 15;        // M row for A, N col for B/D
  const int kg   = lane >> 4;        // K half-group (0/1)
  const int m0   = blockIdx.x * 16;
  const int n0   = wave * 16;

  const float* Xp = X + (size_t)(m0 + row) * D_FEAT + 2 * kg;
  const float* Wp = W + (size_t)(2 * kg) * D_FEAT + n0 + row;

  v8f acc = {};
#pragma unroll
  for (int k0 = 0; k0 < D_FEAT; k0 += 4) {
    v2f a = *(const v2f*)(Xp + k0);                 // contiguous pair along K
    v2f b;
    b.x = Wp[(size_t)k0 * D_FEAT];                  // K rows of W, same column
    b.y = Wp[(size_t)(k0 + 1) * D_FEAT];
    acc = __builtin_amdgcn_wmma_f32_16x16x4_f32(
        /*neg_a=*/false, a, /*neg_b=*/false, b,
        /*c_mod=*/(short)0, acc, /*reuse_a=*/false, /*reuse_b=*/false);
  }

  float* Hp = H + (size_t)(m0 + kg * 8) * D_FEAT + n0 + row;
#pragma unroll
  for (int r = 0; r < 8; ++r) Hp[(size_t)r * D_FEAT] = acc[r];
}

// ------------------------------------------------- message passing (scatter)
// One thread per (edge, 4-feature group): a full wave = one edge, lanes cover
// all 128 features. Gather is a single b128 per thread; atomic targets for a
// wave form one contiguous 512B span at the destination row.
__global__ void k_edge_scatter(const int* __restrict__ ei, int E,
                               const float* __restrict__ h,
                               const float* __restrict__ dinv,
                               float* __restrict__ agg) {
  long long t = (long long)blockIdx.x * blockDim.x + threadIdx.x;
  if (t >= (long long)E * (D_FEAT / 4)) return;
  int e  = (int)(t >> 5);             // D_FEAT/4 == 32 groups per edge
  int f0 = (int)((t & 31) << 2);      // feature base (multiple of 4)
  int s = ei[e];
  int d = ei[E + e];
  float nrm = dinv[s] * dinv[d];
  float4 v = *(const float4*)(h + (size_t)s * D_FEAT + f0);
  float* ap = agg + (size_t)d * D_FEAT + f0;
  atomicAdd(ap + 0, v.x * nrm);
  atomicAdd(ap + 1, v.y * nrm);
  atomicAdd(ap + 2, v.z * nrm);
  atomicAdd(ap + 3, v.w * nrm);
}

// self-loop + bias (+ optional ReLU), in place on agg; float4 streaming
__global__ void k_node_update(float* __restrict__ agg, const float* __restrict__ h,
                              const float* __restrict__ dinv,
                              const float* __restrict__ bias, int N, int relu) {
  long long t = (long long)blockIdx.x * blockDim.x + threadIdx.x;
  if (t >= (long long)N * (D_FEAT / 4)) return;
  int i  = (int)(t >> 5);
  int f0 = (int)((t & 31) << 2);
  float di = dinv[i];
  float dd = di * di;
  size_t base = (size_t)i * D_FEAT + f0;
  float4 a = *(const float4*)(agg + base);
  float4 hv = *(const float4*)(h + base);
  float4 bv = *(const float4*)(bias + f0);
  float4 r;
  r.x = a.x + hv.x * dd + bv.x;
  r.y = a.y + hv.y * dd + bv.y;
  r.z = a.z + hv.z * dd + bv.z;
  r.w = a.w + hv.w * dd + bv.w;
  if (relu) {
    r.x = fmaxf(r.x, 0.0f); r.y = fmaxf(r.y, 0.0f);
    r.z = fmaxf(r.z, 0.0f); r.w = fmaxf(r.w, 0.0f);
  }
  *(float4*)(agg + base) = r;
}

// ---------------------------------------------------------------- mean pool
__global__ void k_pool_accum(const float* __restrict__ z, const int* __restrict__ batch,
                             float* __restrict__ pool, float* __restrict__ cnt, int N) {
  long long t = (long long)blockIdx.x * blockDim.x + threadIdx.x;
  if (t >= (long long)N * (D_FEAT / 4)) return;
  int i  = (int)(t >> 5);
  int f0 = (int)((t & 31) << 2);
  int g = batch[i];
  float4 v = *(const float4*)(z + (size_t)i * D_FEAT + f0);
  float* pp = pool + (size_t)g * D_FEAT + f0;
  atomicAdd(pp + 0, v.x);
  atomicAdd(pp + 1, v.y);
  atomicAdd(pp + 2, v.z);
  atomicAdd(pp + 3, v.w);
  if (f0 == 0) atomicAdd(&cnt[g], 1.0f);
}

__global__ void k_pool_final(const float* __restrict__ pool, const float* __restrict__ cnt,
                             float* __restrict__ out, int G) {
  int t = blockIdx.x * blockDim.x + threadIdx.x;
  if (t >= G * D_FEAT) return;
  int g = t >> 7;
  out[t] = pool[t] / fmaxf(cnt[g], 1.0f);
}

// ---------------------------------------------------------------- launcher
extern "C" void kernel_launch(void* const* d_in, const int* in_sizes, int n_in,
                              void* d_out, int out_size, void* d_ws, size_t ws_size,
                              hipStream_t stream) {
  const float* x     = (const float*)d_in[0];
  const int*   ei    = (const int*)d_in[1];
  const int*   batch = (const int*)d_in[2];
  const float* W1    = (const float*)d_in[3];
  const float* b1    = (const float*)d_in[4];
  const float* W2    = (const float*)d_in[5];
  const float* b2    = (const float*)d_in[6];
  float*       out   = (float*)d_out;

  const int N = in_sizes[0] / D_FEAT;  // 50000 (divisible by 16)
  const int E = in_sizes[1] / 2;       // 640000
  const int G = out_size / D_FEAT;     // 64

  float* ws   = (float*)d_ws;
  float* dinv = ws;
  float* h    = ws + (size_t)((N + 127) & ~127);
  float* agg  = h + (size_t)N * D_FEAT;
  float* pool = agg + (size_t)N * D_FEAT;
  float* cnt  = pool + (size_t)G * D_FEAT;  // contiguous after pool

  const int B = 256;
  auto cdiv = [](long long a, long long b) { return (int)((a + b - 1) / b); };
  const long long NF  = (long long)N * D_FEAT;
  const long long NF4 = NF / 4;
  const long long EF4 = (long long)E * (D_FEAT / 4);

  // degrees -> dinv (in place)
  k_fill_zero<<<cdiv(N, B), B, 0, stream>>>(dinv, N);
  k_deg<<<cdiv(E, B), B, 0, stream>>>(ei + E, dinv, E);
  k_dinv<<<cdiv(N, B), B, 0, stream>>>(dinv, N);

  // ---- layer 1: h = x @ W1 ; agg = norm-scatter(h) + selfloop + b1, ReLU
  k_gemm_wmma<<<N / 16, 256, 0, stream>>>(x, W1, h);
  k_fill_zero<<<cdiv(NF, B), B, 0, stream>>>(agg, (int)NF);
  k_edge_scatter<<<cdiv(EF4, B), B, 0, stream>>>(ei, E, h, dinv, agg);
  k_node_update<<<cdiv(NF4, B), B, 0, stream>>>(agg, h, dinv, b1, N, /*relu=*/1);

  // ---- layer 2: h = z1 @ W2 ; agg = norm-scatter(h) + selfloop + b2
  k_gemm_wmma<<<N / 16, 256, 0, stream>>>(agg, W2, h);
  k_fill_zero<<<cdiv(NF, B), B, 0, stream>>>(agg, (int)NF);
  k_edge_scatter<<<cdiv(EF4, B), B, 0, stream>>>(ei, E, h, dinv, agg);
  k_node_update<<<cdiv(NF4, B), B, 0, stream>>>(agg, h, dinv, b2, N, /*relu=*/0);

  // ---- global mean pool
  k_fill_zero<<<cdiv(G * D_FEAT + G, B), B, 0, stream>>>(pool, G * D_FEAT + G);
  k_pool_accum<<<cdiv(NF4, B), B, 0, stream>>>(agg, batch, pool, cnt, N);
  k_pool_final<<<cdiv((long long)G * D_FEAT, B), B, 0, stream>>>(pool, cnt, out, G);
}